// LearnableSoftLimits6DLayer_38714835206758
// MI455X (gfx1250) — compile-verified
//
#include <hip/hip_runtime.h>
#include <math.h>

// ---------------------------------------------------------------------------
// LearnableSoftLimits6DLayer for MI455X (gfx1250, wave32).
// Memory-bound streaming transform: 6D-rot -> R -> axis/angle -> clamp ->
// Rodrigues -> 6D.  151 MB total traffic -> ~6.5us floor @ 23.3 TB/s.
// Data path: async global->LDS B128 tile staging (ASYNCcnt) + ds_load_b128
// reads + global_store_b128 writes.  VALU trimmed with v_rcp_f32 /
// v_sin_f32 / v_cos_f32 so the kernel stays on the bandwidth roofline.
// No WMMA: no matmul structure exists in this op.
// ---------------------------------------------------------------------------

#ifndef __has_builtin
#define __has_builtin(x) 0
#endif

#if __has_builtin(__builtin_amdgcn_global_load_async_to_lds_b128)
#define HAVE_ASYNC_LDS 1
typedef int v4i __attribute__((ext_vector_type(4)));
typedef __attribute__((address_space(1))) v4i gm_v4i_t;   // global
typedef __attribute__((address_space(3))) v4i lds_v4i_t;  // LDS
#else
#define HAVE_ASYNC_LDS 0
#endif

static __device__ __forceinline__ void wait_async_zero() {
#if HAVE_ASYNC_LDS
#if __has_builtin(__builtin_amdgcn_s_wait_asynccnt)
    __builtin_amdgcn_s_wait_asynccnt(0);
#else
    asm volatile("s_wait_asynccnt 0" ::: "memory");
#endif
#endif
}

// Fast reciprocal: single v_rcp_f32 (~1 ulp), fine next to the EPS guards.
static __device__ __forceinline__ float fast_rcp(float x) {
#if __has_builtin(__builtin_amdgcn_rcpf)
    return __builtin_amdgcn_rcpf(x);
#else
    return 1.0f / x;
#endif
}

constexpr int BLOCK = 256;                 // 8 wave32 waves
constexpr int JPB   = 512;                 // joints per block tile (2/thread)
constexpr int TILE_FLOATS = JPB * 6;       // 3072 floats = 12 KB
constexpr int B128_PER_THREAD = (TILE_FLOATS * 4) / (BLOCK * 16); // = 3
constexpr float EPSF = 1e-6f;

// Faithful per-joint transform (matches reference op order in fp32).
static __device__ __forceinline__ void joint_op(const float* __restrict__ r,
                                                float lo, float hi,
                                                float* __restrict__ o) {
    // --- r6d -> rotmat (Gram-Schmidt), columns b1,b2,b3 ---
    float a1x = r[0], a1y = r[1], a1z = r[2];
    float a2x = r[3], a2y = r[4], a2z = r[5];
    float n1 = sqrtf(fmaf(a1x, a1x, fmaf(a1y, a1y, a1z * a1z)));
    float i1 = fast_rcp(n1 + EPSF);
    float b1x = a1x * i1, b1y = a1y * i1, b1z = a1z * i1;
    float d = fmaf(b1x, a2x, fmaf(b1y, a2y, b1z * a2z));
    float px = fmaf(-d, b1x, a2x);
    float py = fmaf(-d, b1y, a2y);
    float pz = fmaf(-d, b1z, a2z);
    float n2 = sqrtf(fmaf(px, px, fmaf(py, py, pz * pz)));
    float i2 = fast_rcp(n2 + EPSF);
    float b2x = px * i2, b2y = py * i2, b2z = pz * i2;
    float b3x = fmaf(b1y, b2z, -(b1z * b2y));
    float b3y = fmaf(b1z, b2x, -(b1x * b2z));
    float b3z = fmaf(b1x, b2y, -(b1y * b2x));

    // --- rotmat -> axis-angle ---
    float tr   = b1x + b2y + b3z;
    float cosv = fminf(fmaxf((tr - 1.0f) * 0.5f, -1.0f + 1e-7f), 1.0f - 1e-7f);
    float ang  = acosf(cosv);
    float vx = b2z - b3y;            // R21 - R12
    float vy = b3x - b1z;            // R02 - R20
    float vz = b1y - b2x;            // R10 - R01
    float sinv = sqrtf(fmaxf(1.0f - cosv * cosv, 1e-12f));
    float sc   = ang * fast_rcp(2.0f * sinv + EPSF);   // axis * angle fused
    float ax = vx * sc, ay = vy * sc, az = vz * sc;    // aa

    // --- clamp angle magnitude ---
    float ang2 = sqrtf(fmaf(ax, ax, fmaf(ay, ay, az * az)));
    float i3   = fast_rcp(ang2 + EPSF);
    float clamped = fminf(fmaxf(ang2, lo), hi);
    float s2 = clamped * i3;
    float nx = ax * s2, ny = ay * s2, nz = az * s2;    // new_aa

    // --- Rodrigues: R = I + s*K + (1-cos)*K^2, K^2 = xx^T - |x|^2 I ---
    float ang3 = sqrtf(fmaf(nx, nx, fmaf(ny, ny, nz * nz)));
    float i4 = fast_rcp(ang3 + EPSF);
    float x = nx * i4, y = ny * i4, z = nz * i4;
    float s = __sinf(ang3);            // v_sin_f32; |ang3| <~ 3.2 -> accurate
    float c = 1.0f - __cosf(ang3);     // v_cos_f32
    float xy = x * y, xz = x * z, yz = y * z;
    // output = [col0, col1] of R
    o[0] = fmaf(c, -(y * y + z * z), 1.0f);
    o[1] = fmaf(s, z, c * xy);
    o[2] = fmaf(-s, y, c * xz);
    o[3] = fmaf(-s, z, c * xy);
    o[4] = fmaf(c, -(x * x + z * z), 1.0f);
    o[5] = fmaf(s, x, c * yz);
}

__global__ __launch_bounds__(BLOCK)
void soft_limits_6d_kernel(const float* __restrict__ in,
                           const float* __restrict__ min_a,
                           const float* __restrict__ max_a,
                           float* __restrict__ out,
                           long long n_joints, int J) {
    __shared__ float tile[TILE_FLOATS];
    __shared__ float s_lo[128];
    __shared__ float s_hi[128];

    const int tid = threadIdx.x;
    if (tid < J && tid < 128) {
        float lo = min_a[tid];
        s_lo[tid] = lo;
        s_hi[tid] = fmaxf(lo, max_a[tid]);   // hi = max(lo, max_angles)
    }

    const long long tile0 = (long long)blockIdx.x * JPB;
    const bool full_tile = (tile0 + JPB) <= n_joints;   // uniform per block

    if (full_tile) {
        const float* gbase = in + (size_t)tile0 * 6;
#if HAVE_ASYNC_LDS
        // Lane-issued async B128 copies: 3 x 256 x 16B = 12 KB tile (ASYNCcnt).
        #pragma unroll
        for (int k = 0; k < B128_PER_THREAD; ++k) {
            const int f = (k * BLOCK + tid) * 4;        // float index, 16B granular
            __builtin_amdgcn_global_load_async_to_lds_b128(
                (gm_v4i_t*)(gbase + f), (lds_v4i_t*)(&tile[f]),
                /*offset=*/0, /*cpol=*/0);
        }
        wait_async_zero();
#else
        const float4* g4 = reinterpret_cast<const float4*>(gbase);
        float4* t4 = reinterpret_cast<float4*>(tile);
        #pragma unroll
        for (int k = 0; k < B128_PER_THREAD; ++k)
            t4[k * BLOCK + tid] = g4[k * BLOCK + tid];
#endif
        __syncthreads();

        // Each thread owns 2 consecutive joints: 12 floats at 48B stride,
        // 16B-aligned -> 3 x ds_load_b128.
        const float4* L = reinterpret_cast<const float4*>(tile);
        float4 qa = L[tid * 3 + 0];
        float4 qb = L[tid * 3 + 1];
        float4 qc = L[tid * 3 + 2];

        float r0[6] = {qa.x, qa.y, qa.z, qa.w, qb.x, qb.y};
        float r1[6] = {qb.z, qb.w, qc.x, qc.y, qc.z, qc.w};

        const long long g0 = tile0 + (long long)tid * 2;
        const unsigned uJ = (unsigned)J;
        const unsigned jj0 = (unsigned)g0 % uJ;        // one u32 division
        unsigned jj1 = jj0 + 1;                        // derive, no 2nd division
        if (jj1 == uJ) jj1 = 0;

        float o0[6], o1[6];
        joint_op(r0, s_lo[jj0], s_hi[jj0], o0);
        joint_op(r1, s_lo[jj1], s_hi[jj1], o1);

        // 48 contiguous bytes per thread, 16B-aligned (g0 even) -> b128 stores.
        float4* O = reinterpret_cast<float4*>(out + (size_t)g0 * 6);
        O[0] = make_float4(o0[0], o0[1], o0[2], o0[3]);
        O[1] = make_float4(o0[4], o0[5], o1[0], o1[1]);
        O[2] = make_float4(o1[2], o1[3], o1[4], o1[5]);
    } else {
        __syncthreads();   // keep barrier uniform with the full-tile branch
        #pragma unroll
        for (int u = 0; u < 2; ++u) {
            const long long g = tile0 + (long long)tid * 2 + u;
            if (g < n_joints) {
                const float* rp = in + (size_t)g * 6;
                float r[6], o[6];
                #pragma unroll
                for (int i = 0; i < 6; ++i) r[i] = rp[i];
                const unsigned jj = (unsigned)g % (unsigned)J;
                joint_op(r, s_lo[jj], s_hi[jj], o);
                float* op = out + (size_t)g * 6;
                #pragma unroll
                for (int i = 0; i < 6; ++i) op[i] = o[i];
            }
        }
    }
}

extern "C" void kernel_launch(void* const* d_in, const int* in_sizes, int n_in,
                              void* d_out, int out_size, void* d_ws, size_t ws_size,
                              hipStream_t stream) {
    (void)n_in; (void)out_size; (void)d_ws; (void)ws_size;
    const float* pose  = (const float*)d_in[0];   // (n, t, J*6) fp32
    const float* min_a = (const float*)d_in[1];   // (J,) fp32
    const float* max_a = (const float*)d_in[2];   // (J,) fp32
    float* out = (float*)d_out;

    const long long total_floats = (long long)in_sizes[0];
    const int J = in_sizes[1];
    const long long n_joints = total_floats / 6;
    const int tiles = (int)((n_joints + JPB - 1) / JPB);

    soft_limits_6d_kernel<<<tiles, BLOCK, 0, stream>>>(pose, min_a, max_a, out,
                                                       n_joints, J);
}